// AtomicBasis_15685220565082
// MI455X (gfx1250) — compile-verified
//
#include <hip/hip_runtime.h>

// MI455X / gfx1250, wave32. fp32 equivariant message passing.
//  1) mix_wmma_kernel<M>: hm[n][t][c] = sum_j W_r[c][j] * h_r[n][j][m] via
//     V_WMMA_F32_16X16X4_F32 (exact fp32). Done on 25k NODES, not 200k edges
//     (mixing commutes with the gather) -> 8x FLOP reduction + dense GEMM.
//     W_r staged in LDS (bank-conflict-free stride-34 layout).
//  2) zero_kernel: clear d_out (b128 stores).
//  3) edge_msg_kernel: one wave32 per edge, lane = channel. Scalar (SMEM)
//     loads for the wave-uniform edge attrs, 13 coalesced b32 gathers of
//     hm[recv], closed-form contraction math, 13 float atomics into
//     out[send]. hm (41.6MB) and out (41.6MB) are L2-resident (192MB L2).

typedef float v2f __attribute__((ext_vector_type(2)));
typedef float v8f __attribute__((ext_vector_type(8)));

#define NCH 32   // channels
#define NT  13   // 1 + 3 + 9 flattened tensor components
#define WSTRIDE 34  // LDS row stride for W (conflict-free: 34*r distinct mod 64)

// ---------------------------------------------------------------------------
// Kernel 1: channel mixing GEMM via WMMA f32 16x16x4.
// One wave: 32(out-ch) x 16(col) tile. col = n*M + m, M = 3^r (compile-time).
// A = W_r (i x j), from LDS.  B: X[k][col] = h_r[(n*32+k)*M + m], from global.
// ---------------------------------------------------------------------------
template<int M, int TBASE>
__global__ void mix_wmma_kernel(const float* __restrict__ hsrc,
                                const float* __restrict__ W,
                                float* __restrict__ hm,
                                int totalCols, int ntiles)
{
    __shared__ float Wlds[NCH * WSTRIDE];

    // Cooperative LDS fill: 256 threads x 4 floats = 32x32 W slice.
    {
        int t   = threadIdx.x;          // 0..255
        int row = t >> 3;
        int kb  = (t & 7) << 2;
        const float4 wv = *(const float4*)(W + row * 32 + kb);
        float* dst = &Wlds[row * WSTRIDE + kb];
        dst[0] = wv.x; dst[1] = wv.y; dst[2] = wv.z; dst[3] = wv.w;
    }
    __syncthreads();

    int wid = (blockIdx.x * blockDim.x + threadIdx.x) >> 5;
    if (wid >= ntiles) return;          // wave-uniform exit

    int lane     = threadIdx.x & 31;
    int colLocal = lane & 15;
    int koff     = (lane < 16) ? 0 : 2; // A/B k-pair select per lane half

    int  cg    = wid * 16 + colLocal;
    bool valid = cg < totalCols;
    int  cgc   = valid ? cg : (totalCols - 1);  // clamp: loads always in-bounds
    int  n     = cgc / M;                        // M is 1/3/9 -> mul-shift
    int  m     = cgc - n * M;

    // A fragments (16x4 f32) from LDS: lane L<16 -> row i=L, k=q*4+{0,1};
    // lane L>=16 -> row i=L-16, k=q*4+{2,3}. Two M-halves (i 0-15 / 16-31).
    v2f afr[2][8];
#pragma unroll
    for (int hh = 0; hh < 2; ++hh) {
#pragma unroll
        for (int q = 0; q < 8; ++q) {
            const float* ap = &Wlds[(hh * 16 + colLocal) * WSTRIDE + q * 4 + koff];
            v2f a; a.x = ap[0]; a.y = ap[1];    // ds_load_b64
            afr[hh][q] = a;
        }
    }

    // B fragments (4x16 f32): lane holds column col=lane%16, same k pairing.
    v2f bfr[8];
#pragma unroll
    for (int q = 0; q < 8; ++q) {
        int kb = q * 4 + koff;
        v2f b;
        b.x = hsrc[(n * 32 + kb)     * M + m];
        b.y = hsrc[(n * 32 + kb + 1) * M + m];
        bfr[q] = b;
    }

    v8f acc0 = {};   // out channels 0..15
    v8f acc1 = {};   // out channels 16..31
#pragma unroll
    for (int q = 0; q < 8; ++q) {
        acc0 = __builtin_amdgcn_wmma_f32_16x16x4_f32(
            false, afr[0][q], false, bfr[q], (short)0, acc0, false, false);
        acc1 = __builtin_amdgcn_wmma_f32_16x16x4_f32(
            false, afr[1][q], false, bfr[q], (short)0, acc1, false, false);
    }

    // D layout: vgpr v, lanes 0-15 -> row v; lanes 16-31 -> row v+8; col=lane%16.
    // Per lane this is 8 consecutive floats per acc -> 2x global_store_b128.
    if (valid) {
        int g    = (lane >= 16) ? 8 : 0;
        int base = (n * NT + (TBASE + m)) * NCH;
#pragma unroll
        for (int v = 0; v < 8; ++v) {
            hm[base +      (v + g)] = acc0[v];   // i = v+g
            hm[base + 16 + (v + g)] = acc1[v];   // i = 16+v+g
        }
    }
}

// ---------------------------------------------------------------------------
// Kernel 2: zero the output accumulator (vectorized).
// ---------------------------------------------------------------------------
__global__ void zero_kernel(float* __restrict__ p, int n)
{
    int i = (blockIdx.x * blockDim.x + threadIdx.x) * 4;
    if (i + 3 < n) {
        *(float4*)(p + i) = make_float4(0.f, 0.f, 0.f, 0.f);
    } else {
        for (int j = i; j < n; ++j) p[j] = 0.0f;
    }
}

// ---------------------------------------------------------------------------
// Kernel 3: per-edge contraction + scatter. One wave32 per edge, lane=channel.
// ---------------------------------------------------------------------------
__global__ void edge_msg_kernel(const float* __restrict__ hm,
                                const float* __restrict__ ea0,
                                const float* __restrict__ ea1,
                                const float* __restrict__ ea2,
                                const int*   __restrict__ eidx,
                                float* __restrict__ out,
                                int E, int Nn)
{
    int ew = blockIdx.x * 8 + (threadIdx.x >> 5);
    int c  = threadIdx.x & 31;
    int e  = __builtin_amdgcn_readfirstlane(ew);   // SGPR -> scalar loads below
    if (e >= E) return;                            // wave-uniform exit

    int send = eidx[e];                            // edge_index[0][e]  (s_load)
    int recv = eidx[E + e];                        // edge_index[1][e]  (s_load)

    // Edge attributes: uniform index -> scalar (SMEM) loads, 1 fetch per wave
    float a0 = ea0[e];
    float a1[3];
#pragma unroll
    for (int a = 0; a < 3; ++a) a1[a] = ea1[e * 3 + a];
    float A2[3][3];
#pragma unroll
    for (int a = 0; a < 3; ++a)
#pragma unroll
        for (int b = 0; b < 3; ++b) A2[a][b] = ea2[e * 9 + a * 3 + b];

    float trA = A2[0][0] + A2[1][1] + A2[2][2];
    float u   = a0 + trA;                          // a0 + tr(A2)
    float S[3][3];                                 // A2 + A2^T
#pragma unroll
    for (int a = 0; a < 3; ++a)
#pragma unroll
        for (int b = 0; b < 3; ++b) S[a][b] = A2[a][b] + A2[b][a];

    // Gather mixed features at recv (13 fully-coalesced 128B lines)
    const float* hb = hm + (size_t)recv * (NT * NCH);
    float h0 = hb[c];
    float h1[3];
#pragma unroll
    for (int a = 0; a < 3; ++a) h1[a] = hb[(1 + a) * NCH + c];
    float h2[3][3];
#pragma unroll
    for (int a = 0; a < 3; ++a)
#pragma unroll
        for (int b = 0; b < 3; ++b) h2[a][b] = hb[(4 + a * 3 + b) * NCH + c];

    float w = h2[0][0] + h2[1][1] + h2[2][2];      // tr(h2)
    float Hh[3][3];                                 // h2 + h2^T
#pragma unroll
    for (int a = 0; a < 3; ++a)
#pragma unroll
        for (int b = 0; b < 3; ++b) Hh[a][b] = h2[a][b] + h2[b][a];

    // Rank-0: h0*u + h1.a1 + w*u + sum(h2 o S)
    float dot  = h1[0] * a1[0] + h1[1] * a1[1] + h1[2] * a1[2];
    float frob = 0.0f;
#pragma unroll
    for (int a = 0; a < 3; ++a)
#pragma unroll
        for (int b = 0; b < 3; ++b) frob += h2[a][b] * S[a][b];
    float r0 = h0 * u + dot + w * u + frob;

    // Rank-1: h0*a1 + u*h1 + S.h1 + H.a1 + w*a1
    float r1[3];
#pragma unroll
    for (int a = 0; a < 3; ++a) {
        float s = h0 * a1[a] + u * h1[a] + w * a1[a];
#pragma unroll
        for (int b = 0; b < 3; ++b)
            s += S[a][b] * h1[b] + Hh[a][b] * a1[b];
        r1[a] = s;
    }

    // Rank-2: h0*A2 + h1 (x) a1 + u*h2 + (H.S) + w*A2
    float r2[3][3];
#pragma unroll
    for (int a = 0; a < 3; ++a) {
#pragma unroll
        for (int b = 0; b < 3; ++b) {
            float g = 0.0f;
#pragma unroll
            for (int cc = 0; cc < 3; ++cc) g += Hh[a][cc] * S[cc][b];
            r2[a][b] = h0 * A2[a][b] + h1[a] * a1[b] + u * h2[a][b]
                     + g + w * A2[a][b];
        }
    }

    // Scatter-add into out[send] (output = acc0 | acc1 | acc2 concatenated).
    // Per-wave atomic footprint: 128B + 384B + 1152B contiguous spans.
    atomicAdd(&out[send * NCH + c], r0);
    int off1 = Nn * NCH;
#pragma unroll
    for (int a = 0; a < 3; ++a)
        atomicAdd(&out[off1 + (send * NCH + c) * 3 + a], r1[a]);
    int off2 = Nn * NCH * 4;   // Nn*NCH + Nn*NCH*3
#pragma unroll
    for (int a = 0; a < 3; ++a)
#pragma unroll
        for (int b = 0; b < 3; ++b)
            atomicAdd(&out[off2 + (send * NCH + c) * 9 + a * 3 + b], r2[a][b]);
}

// ---------------------------------------------------------------------------
extern "C" void kernel_launch(void* const* d_in, const int* in_sizes, int n_in,
                              void* d_out, int out_size, void* d_ws, size_t ws_size,
                              hipStream_t stream)
{
    const float* h0  = (const float*)d_in[0];
    const float* h1  = (const float*)d_in[1];
    const float* h2  = (const float*)d_in[2];
    // d_in[3] rel_pos: unused by the math
    const float* ea0 = (const float*)d_in[4];
    const float* ea1 = (const float*)d_in[5];
    const float* ea2 = (const float*)d_in[6];
    const float* W   = (const float*)d_in[7];
    const int*   eix = (const int*)d_in[8];

    int Nn = in_sizes[0] / NCH;      // 25000
    int E  = in_sizes[4];            // 200000
    float* hm = (float*)d_ws;        // Nn*13*32 floats = 41.6 MB scratch

    // 1) channel mixing per rank (WMMA GEMM), M = 3^r templated
    {
        int cols, ntiles, blocks;

        cols = Nn * 1;  ntiles = (cols + 15) / 16;  blocks = (ntiles + 7) / 8;
        mix_wmma_kernel<1, 0><<<blocks, 256, 0, stream>>>(
            h0, W + 0 * NCH * NCH, hm, cols, ntiles);

        cols = Nn * 3;  ntiles = (cols + 15) / 16;  blocks = (ntiles + 7) / 8;
        mix_wmma_kernel<3, 1><<<blocks, 256, 0, stream>>>(
            h1, W + 1 * NCH * NCH, hm, cols, ntiles);

        cols = Nn * 9;  ntiles = (cols + 15) / 16;  blocks = (ntiles + 7) / 8;
        mix_wmma_kernel<9, 4><<<blocks, 256, 0, stream>>>(
            h2, W + 2 * NCH * NCH, hm, cols, ntiles);
    }

    // 2) zero the output accumulator
    int zthreads = (out_size + 3) / 4;
    int zblocks  = (zthreads + 255) / 256;
    zero_kernel<<<zblocks, 256, 0, stream>>>((float*)d_out, out_size);

    // 3) per-edge contraction + scatter (one wave per edge)
    int eblocks = (E + 7) / 8;    // 8 waves (edges) per 256-thread block
    edge_msg_kernel<<<eblocks, 256, 0, stream>>>(
        hm, ea0, ea1, ea2, eix, (float*)d_out, E, Nn);
}